// Pfaffian_17532056502830
// MI455X (gfx1250) — compile-verified
//
#include <hip/hip_runtime.h>

// ---------------------------------------------------------------------------
// Pfaffian of a 1024x1024 skew-symmetric matrix (Parlett-Reid, no pivoting),
// blocked for CDNA5: panels of 16 pivot steps, trailing update done as
// A += T*V^T - V*T^T with V_WMMA_F32_16X16X4_F32 (full fp32 — required,
// since the Pfaffian is a product of 512 pivots).
//
// ws layout: A (4 MiB) | T (64 KiB) | V (64 KiB) | idx (4 KiB) | pf (4 B)
// ---------------------------------------------------------------------------

#define NMAT    1024              // Pfaffian matrix dimension (Nparticle)
#define NSITES  1024
#define PANEL   16                // pivot steps per panel = K-dim of update GEMM
#define NPANELS (NMAT / (2 * PANEL))   // 512 steps / 16 = 32 panels

typedef __attribute__((ext_vector_type(2))) float v2f;
typedef __attribute__((ext_vector_type(8))) float v8f;

// ---------------------------------------------------------------------------
__global__ void init_pf(float* pf) {
    if (threadIdx.x == 0) pf[0] = 1.0f;
}

// idx[j] = position of j-th set bit of occ = concat(x>0, x<=0).
// Every site is exactly one of up/down, so occ has exactly NSITES ones:
// ups keep their site index (ranked by prefix count), downs follow at
// total_up + down_rank with value NSITES + site.
__global__ void build_idx(const float* __restrict__ x, int* __restrict__ idx) {
    __shared__ int s[NSITES];
    int t  = threadIdx.x;
    int up = (x[t] > 0.0f) ? 1 : 0;
    s[t] = up;
    __syncthreads();
    // Hillis-Steele inclusive scan over 1024 flags
    for (int off = 1; off < NSITES; off <<= 1) {
        int v = (t >= off) ? s[t - off] : 0;
        __syncthreads();
        s[t] += v;
        __syncthreads();
    }
    int incl     = s[t];
    int excl     = incl - up;          // #up among sites < t
    int total_up = s[NSITES - 1];
    if (up) idx[excl] = t;
    else    idx[total_up + (t - excl)] = NSITES + t;
}

// A[r][c] = F_full[idx[r], idx[c]] with F_full scattered from packed strict
// lower triangle (row-major tril: flat(i,j) = i*(i-1)/2 + j) and antisymmetrized.
__global__ void gather_sub(const float* __restrict__ F, const int* __restrict__ idx,
                           float* __restrict__ A) {
    int c = blockIdx.x * blockDim.x + threadIdx.x;
    int r = blockIdx.y;
    int a = idx[r], b = idx[c];
    float v;
    if (a > b)      v =  F[(long)a * (a - 1) / 2 + b];
    else if (a < b) v = -F[(long)b * (b - 1) / 2 + a];
    else            v = 0.0f;
    A[r * NMAT + c] = v;
}

// Panel factorization: 16 sequential pivot steps. For each step the two pivot
// rows are loaded fresh from A and the pending in-panel rank-2 updates are
// applied in registers (scalars broadcast from LDS). tau/v columns are staged
// in LDS (2*16*1024 f32 = 128 KiB dynamic LDS; gfx1250 WGP has 320 KiB) and
// written out to global T/V for the trailing-update GEMM. The trailing GEMM
// (run afterwards over the whole matrix) re-applies exactly these updates to
// rows i,i+1 in memory, so no write-back of the pivot rows is needed.
__global__ void panel_factor(const float* __restrict__ A,
                             float* __restrict__ Tg, float* __restrict__ Vg,
                             float* __restrict__ pf, int p) {
    extern __shared__ float smem[];
    float* Tl = smem;                      // [PANEL][NMAT]
    float* Vl = smem + PANEL * NMAT;       // [PANEL][NMAT]
    float* bc = smem + 2 * PANEL * NMAT;   // pivot broadcast

    int   l     = threadIdx.x;             // column index, 0..1023
    int   i0    = 2 * PANEL * p;
    float pfacc = 1.0f;

    for (int k = 0; k < PANEL; ++k) {
        int   i   = i0 + 2 * k;
        float ai  = A[i * NMAT + l];
        float ai1 = A[(i + 1) * NMAT + l];
        // pending in-panel updates restricted to the two pivot rows
        for (int m = 0; m < k; ++m) {
            float ti  = Tl[m * NMAT + i];       // broadcast reads
            float vi  = Vl[m * NMAT + i];
            float ti1 = Tl[m * NMAT + i + 1];
            float vi1 = Vl[m * NMAT + i + 1];
            float tl  = Tl[m * NMAT + l];
            float vl  = Vl[m * NMAT + l];
            ai  += ti  * vl - vi  * tl;
            ai1 += ti1 * vl - vi1 * tl;
        }
        if (l == i + 1) bc[0] = ai;
        __syncthreads();
        float pivot = bc[0];
        pfacc *= pivot;
        float tau = (l > i + 1) ? (ai / pivot) : 0.0f;   // masked multipliers
        float vv  = (l > i + 1) ? ai1 : 0.0f;
        Tl[k * NMAT + l] = tau;
        Vl[k * NMAT + l] = vv;
        __syncthreads();   // orders Tl/Vl writes and next step's bc write
    }
    // export panel vectors row-major [NMAT][PANEL] for the WMMA kernel
    for (int m = 0; m < PANEL; ++m) {
        Tg[l * PANEL + m] = Tl[m * NMAT + l];
        Vg[l * PANEL + m] = Vl[m * NMAT + l];
    }
    if (l == 0) pf[0] *= pfacc;   // only one block runs this kernel
}

// Trailing update: A += T*V^T - V*T^T  (n x n x 16), one wave per 16x16 tile.
// V_WMMA_F32_16X16X4_F32, K=16 split into 4 k-chunks per GEMM; the minus is
// folded into the V-operand load (f32 WMMA NEG bits are C-only).
// Layouts (wave32):
//   C/D : vgpr r  <- element (rb + r + 8*(lane>>4), cb + (lane&15))
//   A   : vgpr0/1 <- (M = lane&15, K = kk + 2*(lane>>4) { +0, +1 })
//   B   : vgpr0/1 <- (K = kk + 2*(lane>>4) { +0, +1 }, N = lane&15)
__global__ void trailing_update(float* __restrict__ A,
                                const float* __restrict__ Tg,
                                const float* __restrict__ Vg, int p) {
    int lane = threadIdx.x;                // 0..31
    int rb   = blockIdx.y * 16;
    int cb   = blockIdx.x * 16;
    int lim  = 2 * PANEL * p + 2;          // first index with nonzero tau/v
    if (rb + 16 <= lim || cb + 16 <= lim) return;   // T/V rows all zero there

    int n    = lane & 15;
    int half = lane >> 4;

    v8f c;
#pragma unroll
    for (int r = 0; r < 8; ++r)
        c[r] = A[(rb + r + half * 8) * NMAT + cb + n];

    const float* Ta = Tg + (rb + n) * PANEL;   // A-operand rows of T
    const float* Vb = Vg + (cb + n) * PANEL;   // B-operand: V^T columns
    const float* Va = Vg + (rb + n) * PANEL;   // A-operand rows of (-V)
    const float* Tb = Tg + (cb + n) * PANEL;   // B-operand: T^T columns

#pragma unroll
    for (int kk = 0; kk < PANEL; kk += 4) {    // c += T * V^T
        v2f a, b;
        a.x = Ta[kk + 2 * half];  a.y = Ta[kk + 2 * half + 1];
        b.x = Vb[kk + 2 * half];  b.y = Vb[kk + 2 * half + 1];
        c = __builtin_amdgcn_wmma_f32_16x16x4_f32(false, a, false, b,
                                                  (short)0, c, false, false);
    }
#pragma unroll
    for (int kk = 0; kk < PANEL; kk += 4) {    // c += (-V) * T^T
        v2f a, b;
        a.x = -Va[kk + 2 * half]; a.y = -Va[kk + 2 * half + 1];
        b.x = Tb[kk + 2 * half];  b.y = Tb[kk + 2 * half + 1];
        c = __builtin_amdgcn_wmma_f32_16x16x4_f32(false, a, false, b,
                                                  (short)0, c, false, false);
    }

#pragma unroll
    for (int r = 0; r < 8; ++r)
        A[(rb + r + half * 8) * NMAT + cb + n] = c[r];
}

__global__ void write_out(const float* __restrict__ pf, float* __restrict__ out) {
    if (threadIdx.x == 0) out[0] = pf[0];
}

// ---------------------------------------------------------------------------
extern "C" void kernel_launch(void* const* d_in, const int* in_sizes, int n_in,
                              void* d_out, int out_size, void* d_ws, size_t ws_size,
                              hipStream_t stream) {
    const float* x = (const float*)d_in[0];   // (1024,) spin config
    const float* F = (const float*)d_in[1];   // (1024*2047,) packed tril
    float* out = (float*)d_out;

    char*  ws  = (char*)d_ws;
    float* A   = (float*)ws;                                  // 4 MiB
    float* Tg  = (float*)(ws + (size_t)NMAT * NMAT * sizeof(float));
    float* Vg  = Tg + NMAT * PANEL;
    int*   idx = (int*)(Vg + NMAT * PANEL);
    float* pf  = (float*)(idx + NMAT);

    init_pf<<<1, 32, 0, stream>>>(pf);
    build_idx<<<1, NSITES, 0, stream>>>(x, idx);
    gather_sub<<<dim3(NMAT / 256, NMAT), 256, 0, stream>>>(F, idx, A);

    size_t smem = (size_t)(2 * PANEL * NMAT + 32) * sizeof(float);  // 128 KiB + pad
    for (int p = 0; p < NPANELS; ++p) {
        panel_factor<<<1, NMAT, smem, stream>>>(A, Tg, Vg, pf, p);
        trailing_update<<<dim3(NMAT / 16, NMAT / 16), 32, 0, stream>>>(A, Tg, Vg, p);
    }
    write_out<<<1, 32, 0, stream>>>(pf, out);
}